// APPNPXSimp_BN_55121610277364
// MI455X (gfx1250) — compile-verified
//
#include <hip/hip_runtime.h>

// Problem constants (match reference)
#define IN_F   500
#define HID    64
#define CLS    40
#define ALPHA  0.1f
#define KPROP  10
#define BN_EPS 1e-5f

typedef float v2f __attribute__((ext_vector_type(2)));
typedef float v4f __attribute__((ext_vector_type(4)));
typedef float v8f __attribute__((ext_vector_type(8)));
typedef unsigned int u32x4v __attribute__((ext_vector_type(4)));
typedef int i32x4v __attribute__((ext_vector_type(4)));
typedef int i32x8v __attribute__((ext_vector_type(8)));

#if __has_include(<hip/amd_detail/amd_gfx1250_TDM.h>)
#define TDM_6ARG 1
#endif

// ---------------------------------------------------------------- utilities
__global__ void fill_f32(float* __restrict__ p, float v, long long n) {
  long long i = (long long)blockIdx.x * blockDim.x + threadIdx.x;
  if (i < n) p[i] = v;
}

// deg[dst] += 1 per edge (deg pre-filled with 1.0 for the self loop)
__global__ void deg_count(const int* __restrict__ dst, float* __restrict__ deg, int E) {
  int e = blockIdx.x * blockDim.x + threadIdx.x;
  if (e < E) atomicAdd(&deg[dst[e]], 1.0f);
}

// dinv = rsqrt(deg) (deg >= 1 due to self loops); sn = dinv^2 (self-loop norm)
__global__ void dinv_kernel(const float* __restrict__ deg, float* __restrict__ dinv,
                            float* __restrict__ sn, int n) {
  int i = blockIdx.x * blockDim.x + threadIdx.x;
  if (i < n) { float r = rsqrtf(deg[i]); dinv[i] = r; sn[i] = r * r; }
}

__global__ void edge_norm(const int* __restrict__ src, const int* __restrict__ dst,
                          const float* __restrict__ dinv, float* __restrict__ norm, int E) {
  int e = blockIdx.x * blockDim.x + threadIdx.x;
  if (e < E) norm[e] = dinv[src[e]] * dinv[dst[e]];
}

// Swizzle B [K x Nc] into fragment-major layout Bs[(K/4)][lane(32)][2*NT]:
//   j = 2*t + p  ->  B[row = 4*kb + 2*(lane>>4) + p][col = 16*t + (lane&15)]
// so each lane's per-4-kstep B data is contiguous (vector loadable), zero-padded.
__global__ void swizzle_B(const float* __restrict__ B, float* __restrict__ Bs,
                          int K, int Nc, int NT) {
  const int j2 = 2 * NT;
  const int total = (K / 4) * 32 * j2;
  int idx = blockIdx.x * blockDim.x + threadIdx.x;
  if (idx >= total) return;
  int j    = idx % j2;
  int lane = (idx / j2) % 32;
  int kb   = idx / (j2 * 32);
  int row  = kb * 4 + 2 * (lane >> 4) + (j & 1);
  int col  = (j >> 1) * 16 + (lane & 15);
  Bs[idx] = (col < Nc) ? B[row * Nc + col] : 0.0f;
}

// ------------------------------------------------------------ WMMA f32 GEMM
// C[M x Nc] = A[M x K] * B + bias, fp32 exact via V_WMMA_F32_16X16X4_F32.
// One wave32 owns one 16-row tile and ALL NT column tiles (A-fragment reuse).
// The swizzled B panel ((K/4)*32*2*NT floats, <= 125 KB) is staged into LDS by
// the Tensor Data Mover (wave 0 issues tensor_load_to_lds, s_wait_tensorcnt,
// workgroup barrier), then the inner loop reads B with aligned ds_load vectors.
// Requires: M % 16 == 0, K % 4 == 0.
template <int NT>
__global__ __launch_bounds__(256)
void wmma_gemm_f32(const float* __restrict__ A, const float* __restrict__ Bs,
                   const float* __restrict__ bias, float* __restrict__ C,
                   int M, int K, int Nc) {
  extern __shared__ float sB[];
  const int lane = threadIdx.x & 31;
  const int wave = threadIdx.x >> 5;
  const unsigned int totalF = (unsigned)(K / 4) * 32u * (unsigned)(2 * NT);

  // ---- stage swizzled B panel into LDS
#if __has_builtin(__builtin_amdgcn_tensor_load_to_lds)
  if (wave == 0) {                                  // one TDM issue per block
    const unsigned long long ga = (unsigned long long)(size_t)Bs;
    const unsigned int ldsAddr  = (unsigned int)(size_t)(void*)sB;
    u32x4v g0;
    g0.x = 1u;                                      // count=1 (valid user D#)
    g0.y = ldsAddr;                                 // lds_addr (bytes)
    g0.z = (unsigned int)(ga & 0xffffffffu);        // global_addr[31:0]
    g0.w = (unsigned int)((ga >> 32) & 0x01ffffffu) // global_addr[56:32]
         | (2u << 30);                              // type=2 ("image")
    i32x8v g1 = {};
    g1[0] = (int)(2u << 16);                        // data_size=4B, wg_mask=0
    g1[1] = (int)((totalF & 0xffffu) << 16);        // tensor_dim0[15:0]
    g1[2] = (int)((totalF >> 16) | (1u << 16));     // tensor_dim0[31:16], tensor_dim1=1
    g1[3] = (int)((totalF & 0xffffu) << 16);        // tile_dim0 = totalF (<= 65535)
    g1[4] = 1;                                      // tile_dim1=1, tile_dim2=0
    g1[5] = (int)totalF;                            // tensor_dim0_stride[31:0]
    g1[6] = 0;
    g1[7] = 0;
    i32x4v g2 = {};
    i32x4v g3 = {};
#ifdef TDM_6ARG
    __builtin_amdgcn_tensor_load_to_lds(g0, g1, g2, g3, (i32x8v){}, 0);
#else
    __builtin_amdgcn_tensor_load_to_lds(g0, g1, g2, g3, 0);
#endif
    __builtin_amdgcn_s_wait_tensorcnt(0);
  }
#else
  for (unsigned int i = threadIdx.x; i < totalF; i += blockDim.x) sB[i] = Bs[i];
#endif
  __syncthreads();

  const int mt = blockIdx.x * (blockDim.x >> 5) + wave;
  if (mt * 16 < M) {
    const int half = lane >> 4;                     // 0: K={k,k+1}, 1: K={k+2,k+3}
    const int l15  = lane & 15;

    const float* Ap = A + (size_t)(mt * 16 + l15) * K + half * 2;
    const float* Bl = sB + lane * (2 * NT);
    const int bstride = 32 * 2 * NT;                // floats per 4-k block

    v8f acc[NT];
#pragma unroll
    for (int t = 0; t < NT; ++t) acc[t] = (v8f){};

    for (int k = 0; k < K; k += 4) {
      const v2f a = *(const v2f*)Ap;                // global_load_b64, K={ka,ka+1}
      __builtin_prefetch(Ap + 32, 0, 1);            // stream A ahead
      float bl[2 * NT];
      if constexpr (NT == 4) {                      // 32B-aligned: 2x ds_load_b128
        *(v4f*)&bl[0] = *(const v4f*)(Bl);
        *(v4f*)&bl[4] = *(const v4f*)(Bl + 4);
      } else {                                      // 8B-aligned: 3x ds_load_b64
        *(v2f*)&bl[0] = *(const v2f*)(Bl);
        *(v2f*)&bl[2] = *(const v2f*)(Bl + 2);
        *(v2f*)&bl[4] = *(const v2f*)(Bl + 4);
      }
#pragma unroll
      for (int t = 0; t < NT; ++t) {
        v2f b;
        b.x = bl[2 * t];                            // row ka,   col t*16+l15
        b.y = bl[2 * t + 1];                        // row ka+1, col t*16+l15
        acc[t] = __builtin_amdgcn_wmma_f32_16x16x4_f32(
            /*neg_a=*/false, a, /*neg_b=*/false, b,
            /*c_mod=*/(short)0, acc[t], /*reuse_a=*/false, /*reuse_b=*/false);
      }
      Ap += 4;
      Bl += bstride;
    }

#pragma unroll
    for (int t = 0; t < NT; ++t) {
      const int col = t * 16 + l15;
      if (col < Nc) {
        const float bv = bias[col];
#pragma unroll
        for (int v = 0; v < 8; ++v) {
          const int row = mt * 16 + v + half * 8;
          C[(size_t)row * Nc + col] = acc[t][v] + bv;
        }
      }
    }
  }
}

// --------------------------------------------------------------- APPNP step
// agg[dst, 2f..2f+1] += h[src, 2f..2f+1] * norm[e]; blockDim = (F/2, edges_per_block)
__global__ void scatter_edges(const float* __restrict__ h, const int* __restrict__ src,
                              const int* __restrict__ dst, const float* __restrict__ norm,
                              float* __restrict__ agg, int E, int F) {
  const int e = blockIdx.x * blockDim.y + threadIdx.y;
  const int f = threadIdx.x * 2;
  if (e >= E) return;
  const int s = src[e];
  const int d = dst[e];
  const float nv = norm[e];
  const v2f hv = *(const v2f*)(h + (size_t)s * F + f);   // global_load_b64
  float* ap = agg + (size_t)d * F + f;
  atomicAdd(ap + 0, hv.x * nv);
  atomicAdd(ap + 1, hv.y * nv);
}

// out = (1-alpha)*(agg + cur*selfnorm) + alpha*x0
__global__ void appnp_combine(const float* __restrict__ agg, const float* __restrict__ cur,
                              const float* __restrict__ x0, const float* __restrict__ sn,
                              float* __restrict__ out, int n, int F) {
  const int idx = blockIdx.x * blockDim.x + threadIdx.x;
  if (idx >= n * F) return;
  const int r = idx / F;
  const float a = agg[idx] + cur[idx] * sn[r];
  out[idx] = (1.0f - ALPHA) * a + ALPHA * x0[idx];
}

// ------------------------------------------------------------- BatchNorm1d
__global__ void bn_stats(const float* __restrict__ x, float* __restrict__ gsum,
                         float* __restrict__ gsq, int n, int F) {
  __shared__ float ssum[64];
  __shared__ float ssq[64];
  const int tid = threadIdx.x;
  if (tid < F) { ssum[tid] = 0.0f; ssq[tid] = 0.0f; }
  __syncthreads();
  const int total = n * F;
  for (int idx = blockIdx.x * blockDim.x + tid; idx < total; idx += gridDim.x * blockDim.x) {
    const float v = x[idx];
    const int f = idx % F;
    atomicAdd(&ssum[f], v);                         // ds_add_f32
    atomicAdd(&ssq[f], v * v);
  }
  __syncthreads();
  if (tid < F) {
    atomicAdd(&gsum[tid], ssum[tid]);
    atomicAdd(&gsq[tid], ssq[tid]);
  }
}

__global__ void bn_apply(const float* __restrict__ x, const float* __restrict__ gsum,
                         const float* __restrict__ gsq, const float* __restrict__ g,
                         const float* __restrict__ b, float* __restrict__ out,
                         int n, int F, int do_relu) {
  const int idx = blockIdx.x * blockDim.x + threadIdx.x;
  if (idx >= n * F) return;
  const int f = idx % F;
  const float invn = 1.0f / (float)n;
  const float mu   = gsum[f] * invn;
  const float var  = gsq[f] * invn - mu * mu;
  const float s    = g[f] * rsqrtf(var + BN_EPS);
  float v = (x[idx] - mu) * s + b[f];
  if (do_relu) v = fmaxf(v, 0.0f);
  out[idx] = v;
}

// ------------------------------------------------------------------- driver
static inline int cdiv_i(long long a, int b) { return (int)((a + b - 1) / b); }

extern "C" void kernel_launch(void* const* d_in, const int* in_sizes, int n_in,
                              void* d_out, int out_size, void* d_ws, size_t ws_size,
                              hipStream_t stream) {
  (void)n_in; (void)out_size; (void)ws_size;

  const float* x   = (const float*)d_in[0];
  const int*   ei  = (const int*)d_in[1];        // [2, E] int32
  const float* W1  = (const float*)d_in[2];
  const float* b1  = (const float*)d_in[3];
  const float* Wm  = (const float*)d_in[4];
  const float* bm  = (const float*)d_in[5];
  const float* W2  = (const float*)d_in[6];
  const float* b2  = (const float*)d_in[7];
  const float* g1  = (const float*)d_in[8];
  const float* be1 = (const float*)d_in[9];
  const float* g3  = (const float*)d_in[10];
  const float* be3 = (const float*)d_in[11];
  const float* g2  = (const float*)d_in[12];
  const float* be2 = (const float*)d_in[13];

  const int E = in_sizes[1] / 2;
  const int n = in_sizes[0] / IN_F;              // 100000 (multiple of 16)
  const int* src = ei;
  const int* dst = ei + E;

  // ---- carve workspace (256B aligned regions)
  char* wp = (char*)d_ws;
  auto carve = [&](size_t bytes) -> float* {
    float* r = (float*)wp;
    wp += (bytes + 255) & ~(size_t)255;
    return r;
  };
  float* deg    = carve((size_t)n * 4);
  float* dinv   = carve((size_t)n * 4);
  float* sn     = carve((size_t)n * 4);
  float* norm   = carve((size_t)E * 4);
  float* h0     = carve((size_t)n * HID * 4);    // also reused as 40-wide logits
  float* hA     = carve((size_t)n * HID * 4);
  float* hB     = carve((size_t)n * HID * 4);
  float* agg    = carve((size_t)n * HID * 4);
  float* Bs     = carve((size_t)(IN_F / 4) * 32 * 8 * 4);  // largest swizzled panel
  float* bstats = carve(128 * 4);
  float* bsum = bstats;
  float* bsq  = bstats + 64;

  // ---- GCN normalization
  fill_f32<<<cdiv_i(n, 256), 256, 0, stream>>>(deg, 1.0f, n);   // self loop contributes 1
  deg_count<<<cdiv_i(E, 256), 256, 0, stream>>>(dst, deg, E);
  dinv_kernel<<<cdiv_i(n, 256), 256, 0, stream>>>(deg, dinv, sn, n);
  edge_norm<<<cdiv_i(E, 256), 256, 0, stream>>>(src, dst, dinv, norm, E);

  auto gemm = [&](const float* A, const float* B, const float* bias, float* Cout,
                  int M, int K, int Nc) {
    const int ntn = (Nc + 15) / 16;              // 4 (H=64) or 3 (C=40)
    const long long panel = (long long)(K / 4) * 32 * 2 * ntn;     // floats
    swizzle_B<<<cdiv_i(panel, 256), 256, 0, stream>>>(B, Bs, K, Nc, ntn);
    const int blocks = cdiv_i(M / 16, 8);        // 8 waves/block, 1 row-tile/wave
    const size_t sbytes = (size_t)panel * 4;     // dynamic LDS (<= 125 KB < 320 KB)
    if (ntn == 4)
      wmma_gemm_f32<4><<<blocks, 256, sbytes, stream>>>(A, Bs, bias, Cout, M, K, Nc);
    else
      wmma_gemm_f32<3><<<blocks, 256, sbytes, stream>>>(A, Bs, bias, Cout, M, K, Nc);
  };

  // K propagation steps; returns pointer to buffer holding the result
  auto appnp = [&](const float* x0, float* bufA, float* bufB, int F) -> float* {
    const float* cur = x0;
    float* nxt = bufA;
    float* other = bufB;
    const int half = F / 2;                      // 32 (H) or 20 (C)
    const dim3 sblk(half, 256 / half);           // 32x8 or 20x12 threads
    const int sgrid = cdiv_i(E, sblk.y);
    const int cgrid = cdiv_i((long long)n * F, 256);
    for (int it = 0; it < KPROP; ++it) {
      fill_f32<<<cgrid, 256, 0, stream>>>(agg, 0.0f, (long long)n * F);
      scatter_edges<<<sgrid, sblk, 0, stream>>>(cur, src, dst, norm, agg, E, F);
      appnp_combine<<<cgrid, 256, 0, stream>>>(agg, cur, x0, sn, nxt, n, F);
      cur = nxt;
      float* t = nxt; nxt = other; other = t;
    }
    return (float*)cur;
  };

  auto batchnorm = [&](const float* in, const float* g, const float* b, float* out,
                       int F, int relu) {
    fill_f32<<<1, 128, 0, stream>>>(bstats, 0.0f, 128);
    bn_stats<<<512, 256, 0, stream>>>(in, bsum, bsq, n, F);
    bn_apply<<<cdiv_i((long long)n * F, 256), 256, 0, stream>>>(in, bsum, bsq, g, b, out,
                                                                n, F, relu);
  };

  // ---- layer 1: x @ W1 + b1 -> APPNP -> BN+ReLU
  gemm(x, W1, b1, h0, n, IN_F, HID);
  float* r1 = appnp(h0, hA, hB, HID);
  float* o1 = (r1 == hA) ? hB : hA;
  batchnorm(r1, g1, be1, o1, HID, 1);

  // ---- middle block: h @ Wm + bm -> APPNP -> BN+ReLU
  gemm(o1, Wm, bm, h0, n, HID, HID);
  float* r2 = appnp(h0, hA, hB, HID);
  float* o2 = (r2 == hA) ? hB : hA;
  batchnorm(r2, g3, be3, o2, HID, 1);

  // ---- output: h @ W2 + b2 -> APPNP -> BN (no relu)
  gemm(o2, W2, b2, h0, n, HID, CLS);             // h0 reused as [n, 40]
  float* r3 = appnp(h0, hA, hB, CLS);
  batchnorm(r3, g2, be2, (float*)d_out, CLS, 0);
}